// VertexToImage_51556787421215
// MI455X (gfx1250) — compile-verified
//
#include <hip/hip_runtime.h>
#include <hip/hip_bf16.h>
#include <stdint.h>

// out[l][k][d] = x[l + k - 7][d]  (zero if source row out of [0, L))
// L = 8192, D = 1024 floats (4096 B per row), w = 15.
//
// Pure data movement (~0.5 GiB total traffic; ~22 us at 23.3 TB/s):
//   - one block per input row; async-load the 4 KiB row into LDS once
//     (input read from HBM exactly once, 32 MiB total)
//   - fan it out to <=15 output rows with async NT stores from LDS
//     (480 MiB streamed, never touching VGPRs, pipelined behind ASYNCcnt)
//   - boundary zero rows handled by a tiny separate kernel

#define D_FLOATS   1024
#define ROW_BYTES  4096u                 // D_FLOATS * 4
#define PAD        7
#define W          15
#define DST_STEP   57344ull              // 14 * ROW_BYTES  (address delta per k)
#define DST_SPAN   61440ull              // 15 * ROW_BYTES

__global__ void __launch_bounds__(256)
window_copy_async(const float* __restrict__ x, float* __restrict__ out, int L) {
    const int r = blockIdx.x;            // input row index
    const int t = threadIdx.x;           // 0..255, each lane owns 16 bytes of the row

    // Dynamic LDS (4096 B passed at launch) -> allocation is guaranteed by the
    // dispatch packet even though only inline asm touches it. Base offset 0.
    extern __shared__ __align__(16) char smem[];
    const uint32_t lds_addr = (uint32_t)(uintptr_t)smem + (uint32_t)t * 16u;

    // Async copy: global row r -> LDS (per-lane 16B), tracked by ASYNCcnt.
    const uint64_t gsrc = (uint64_t)(uintptr_t)x
                        + (uint64_t)r * ROW_BYTES + (uint64_t)t * 16u;
    asm volatile("global_load_async_to_lds_b128 %0, %1, off"
                 :: "v"(lds_addr), "v"(gsrc)
                 : "memory");

    // Wait for this wave's async load to land in LDS. Each wave stores only
    // the LDS bytes it loaded itself, so no workgroup barrier is required.
    asm volatile("s_wait_asynccnt 0" ::: "memory");

    // Row r lands at out[l][k] for l = r + 7 - k, k = 0..14.
    // Flat dst offset: (l*15 + k)*4096 = (r+7)*61440 - k*57344.
    const uint64_t a0 = (uint64_t)(uintptr_t)out
                      + (uint64_t)(r + PAD) * DST_SPAN + (uint64_t)t * 16u;

    if (r >= PAD && r < L - PAD) {
        // Interior rows (8178 of 8192): all 15 destinations valid.
        uint64_t a = a0;
#pragma unroll
        for (int k = 0; k < W; ++k) {
            asm volatile("global_store_async_from_lds_b128 %0, %1, off th:TH_STORE_NT"
                         :: "v"(a), "v"(lds_addr)
                         : "memory");
            a -= DST_STEP;
        }
    } else {
        // Boundary rows: guard each destination. Condition is uniform per
        // block (depends only on blockIdx) -> EXEC stays all-ones.
#pragma unroll
        for (int k = 0; k < W; ++k) {
            const int l = r + PAD - k;
            if (0 <= l && l < L) {
                const uint64_t a = a0 - (uint64_t)k * DST_STEP;
                asm volatile("global_store_async_from_lds_b128 %0, %1, off th:TH_STORE_NT"
                             :: "v"(a), "v"(lds_addr)
                             : "memory");
            }
        }
    }
    // S_ENDPGM performs an implicit wait-idle which drains ASYNCcnt.
}

// Zero the 56 boundary rows (source row outside [0, L)).
// grid = 2 * PAD * W = 210 blocks of 256 threads; inactive blocks return.
__global__ void __launch_bounds__(256)
window_zero(float4* __restrict__ out, int L) {
    const int b    = blockIdx.x;
    const int half = b / (PAD * W);      // 0 = head (l < 7), 1 = tail (l >= L-7)
    const int idx  = b % (PAD * W);
    const int ll   = idx / W;
    const int k    = idx % W;
    const int l    = half ? (L - PAD + ll) : ll;
    const int src  = l + k - PAD;
    if (src >= 0 && src < L) return;     // in-range rows handled by copy kernel

    const float4 z = make_float4(0.f, 0.f, 0.f, 0.f);
    const size_t base = ((size_t)l * W + (size_t)k) * (D_FLOATS / 4);
    out[base + threadIdx.x] = z;
}

extern "C" void kernel_launch(void* const* d_in, const int* in_sizes, int n_in,
                              void* d_out, int out_size, void* d_ws, size_t ws_size,
                              hipStream_t stream) {
    const float* x = (const float*)d_in[0];
    float* out = (float*)d_out;
    const int L = in_sizes[0] / D_FLOATS;   // 8192

    window_zero<<<2 * PAD * W, 256, 0, stream>>>((float4*)out, L);
    window_copy_async<<<L, 256, ROW_BYTES, stream>>>(x, out, L);
}